// MemoryAttentionConv_79242146611370
// MI455X (gfx1250) — compile-verified
//
#include <hip/hip_runtime.h>
#include <hip/hip_bf16.h>

// Sizes fixed by the reference
#define Bsz   4
#define CINc  128
#define CHALF 64
#define Nn    4096
#define Kk    20
#define Ll    4
#define COUTc 64
#define NPOS  (Nn * Kk)          // 81920 columns per (b, channel)

typedef float v2f __attribute__((ext_vector_type(2)));
typedef float v4f __attribute__((ext_vector_type(4)));
typedef float v8f __attribute__((ext_vector_type(8)));

// ---------------------------------------------------------------------------
// Kernel 1: q = Wq*abs_x, kq = Wk*x, vq = Wv*x  via V_WMMA_F32_16X16X4_F32
// Grid: (NPOS/64, B), block 128 (4 waves). Wave w owns output rows [16w,16w+16).
// x tile (128x64) and abs_x tile (64x64) staged in LDS with b128 loads/stores;
// weights preloaded in registers (L2-resident, 80KB total).
// ---------------------------------------------------------------------------
__global__ __launch_bounds__(128)
void gemm_qkv_kernel(const float* __restrict__ x, const float* __restrict__ abs_x,
                     const float* __restrict__ Wq, const float* __restrict__ Wk,
                     const float* __restrict__ Wv,
                     float* __restrict__ kq, float* __restrict__ vq,
                     float* __restrict__ qq)
{
    __shared__ float lds_x[CINc * 64];    // 32 KB
    __shared__ float lds_ax[CHALF * 64];  // 16 KB

    const int b     = blockIdx.y;
    const int col0  = blockIdx.x * 64;
    const int tid   = threadIdx.x;
    const int wave  = tid >> 5;
    const int lane  = tid & 31;
    const int lhalf = lane & 15;
    const int koff  = (lane >> 4) * 2;    // K offset of this lane-half (A & B frags)

    // --- stage x tile: 128 rows x 64 cols, vectorized float4 (b128) ---
    // 2048 float4 elements; 16 float4 per row. Consecutive threads -> consecutive
    // 16B chunks of one row: fully coalesced global_load_b128 / ds_store_b128.
    const float* xb = x + (size_t)b * CINc * NPOS;
    #pragma unroll
    for (int i = 0; i < 16; ++i) {
        int ev = tid + i * 128;           // 0..2047
        int r = ev >> 4, c4 = ev & 15;
        ((v4f*)lds_x)[ev] = *(const v4f*)(xb + (size_t)r * NPOS + col0 + c4 * 4);
    }
    const float* axb = abs_x + (size_t)b * CHALF * NPOS;
    #pragma unroll
    for (int i = 0; i < 8; ++i) {
        int ev = tid + i * 128;           // 0..1023
        int r = ev >> 4, c4 = ev & 15;
        ((v4f*)lds_ax)[ev] = *(const v4f*)(axb + (size_t)r * NPOS + col0 + c4 * 4);
    }
    __syncthreads();

    const int m0 = wave * 16;             // output-row tile of this wave

    // --- preload A fragments (weights) into registers ---
    // A 16x4 f32 layout: lanes 0-15 -> M=lane, K={0,1}; lanes 16-31 -> K={2,3}
    v2f ak[32], av[32], aq[16];
    #pragma unroll
    for (int t2 = 0; t2 < 32; ++t2) {
        int kkb = t2 * 4 + koff;
        ak[t2].x = Wk[(m0 + lhalf) * CINc + kkb];
        ak[t2].y = Wk[(m0 + lhalf) * CINc + kkb + 1];
        av[t2].x = Wv[(m0 + lhalf) * CINc + kkb];
        av[t2].y = Wv[(m0 + lhalf) * CINc + kkb + 1];
    }
    #pragma unroll
    for (int t2 = 0; t2 < 16; ++t2) {
        int kkb = t2 * 4 + koff;
        aq[t2].x = Wq[(m0 + lhalf) * CHALF + kkb];
        aq[t2].y = Wq[(m0 + lhalf) * CHALF + kkb + 1];
    }

    for (int nt = 0; nt < 4; ++nt) {
        const int c0 = nt * 16;
        v8f acck = {}; v8f accv = {}; v8f accq = {};

        // B 4x16 f32 layout: lanes 0-15 -> N=lane, K={0,1}; lanes 16-31 -> K={2,3}
        #pragma unroll
        for (int t2 = 0; t2 < 32; ++t2) {
            int kkb = t2 * 4 + koff;
            v2f bf;
            bf.x = lds_x[(kkb    ) * 64 + c0 + lhalf];
            bf.y = lds_x[(kkb + 1) * 64 + c0 + lhalf];
            acck = __builtin_amdgcn_wmma_f32_16x16x4_f32(false, ak[t2], false, bf,
                                                         (short)0, acck, false, false);
            accv = __builtin_amdgcn_wmma_f32_16x16x4_f32(false, av[t2], false, bf,
                                                         (short)0, accv, false, false);
        }
        #pragma unroll
        for (int t2 = 0; t2 < 16; ++t2) {
            int kkb = t2 * 4 + koff;
            v2f bf;
            bf.x = lds_ax[(kkb    ) * 64 + c0 + lhalf];
            bf.y = lds_ax[(kkb + 1) * 64 + c0 + lhalf];
            accq = __builtin_amdgcn_wmma_f32_16x16x4_f32(false, aq[t2], false, bf,
                                                         (short)0, accq, false, false);
        }

        // C/D layout: VGPR i -> M = i (lanes 0-15) or i+8 (lanes 16-31), N = lhalf
        const int colg  = col0 + c0 + lhalf;
        const int rbase = m0 + ((lane >> 4) << 3);
        #pragma unroll
        for (int i = 0; i < 8; ++i) {
            size_t off = ((size_t)(b * COUTc + rbase + i)) * NPOS + colg;
            kq[off] = acck[i];
            vq[off] = accv[i];
            qq[off] = accq[i];
        }
    }
}

// ---------------------------------------------------------------------------
// Kernel 2: out_q = sum_k softmax(q*kq) * vq ; segment-sum of weights -> score
// Grid: (N/4, B), block 256: thread = (ln = t/64 -> n, c = t%64).
// ---------------------------------------------------------------------------
__global__ __launch_bounds__(256)
void attn_query_kernel(const float* __restrict__ qq, const float* __restrict__ kq,
                       const float* __restrict__ vq, const int* __restrict__ idx,
                       float* __restrict__ out, float* __restrict__ score)
{
    __shared__ int sidx[4 * Kk];
    const int b  = blockIdx.y;
    const int n0 = blockIdx.x * 4;
    const int t  = threadIdx.x;
    const int ln = t >> 6;
    const int c  = t & 63;
    const int n  = n0 + ln;

    if (t < 4 * Kk) {                     // idx[b, n, k, L] (last of L+1)
        int lln = t / Kk, k2 = t % Kk;
        sidx[t] = idx[(((size_t)b * Nn + n0 + lln) * Kk + k2) * (Ll + 1) + Ll];
    }
    __syncthreads();

    const size_t base = ((size_t)(b * COUTc + c)) * NPOS + (size_t)n * Kk;
    float m[Kk], vv[Kk];
    float mx = -__builtin_inff();
    #pragma unroll
    for (int k2 = 0; k2 < Kk; ++k2) {
        float qv = qq[base + k2];
        float kv = kq[base + k2];
        vv[k2] = vq[base + k2];
        m[k2] = qv * kv;
        mx = fmaxf(mx, m[k2]);
    }
    float ssum = 0.f;
    #pragma unroll
    for (int k2 = 0; k2 < Kk; ++k2) { m[k2] = __expf(m[k2] - mx); ssum += m[k2]; }
    const float inv = 1.0f / ssum;

    float o = 0.f;
    const size_t srow = ((size_t)(b * COUTc + c)) * Nn;
    #pragma unroll
    for (int k2 = 0; k2 < Kk; ++k2) {
        float w = m[k2] * inv;
        o += w * vv[k2];
        atomicAdd(&score[srow + sidx[ln * Kk + k2]], w);
    }
    out[((size_t)(b * 2 * COUTc + c)) * Nn + n] = o;
}

// ---------------------------------------------------------------------------
// Kernel 3: memory attention. One thread per (b,c,n):
//   gather 80 scores, stable top-20 (desc, ties -> lower index), gather k/v at
//   selected indices, softmax(q * k_sel) . v_sel -> out[:, 64+c, :].
// ---------------------------------------------------------------------------
__global__ __launch_bounds__(256)
void attn_mem_kernel(const float* __restrict__ qq, const float* __restrict__ kmem,
                     const float* __restrict__ vmem, const float* __restrict__ s,
                     const int* __restrict__ idx, float* __restrict__ out)
{
    const int gid = blockIdx.x * 256 + threadIdx.x;   // < B*COUT*N = 1,048,576
    const int n = gid & (Nn - 1);
    const int c = (gid >> 12) & 63;
    const int b = gid >> 18;

    float topv[20]; int topi[20];
    #pragma unroll
    for (int j = 0; j < 20; ++j) { topv[j] = -__builtin_inff(); topi[j] = 0; }

    const size_t idxbase = ((size_t)b * Nn + n) * (Kk * (Ll + 1));
    const size_t sbase   = ((size_t)(b * COUTc + c)) * Nn;
    #pragma unroll
    for (int k2 = 0; k2 < Kk; ++k2) {
        #pragma unroll
        for (int l = 0; l < Ll; ++l) {
            int   si  = idx[idxbase + k2 * (Ll + 1) + l];
            float val = s[((size_t)(sbase + si)) * Ll + l];
            if (val > topv[19]) {                  // strict: equal keeps earlier idx
                topv[19] = val; topi[19] = k2 * Ll + l;
                #pragma unroll
                for (int p = 19; p > 0; --p) {     // bubble-up while strictly greater
                    if (topv[p] > topv[p - 1]) {
                        float tv = topv[p]; topv[p] = topv[p - 1]; topv[p - 1] = tv;
                        int   ti = topi[p]; topi[p] = topi[p - 1]; topi[p - 1] = ti;
                    }
                }
            }
        }
    }

    const size_t qbase  = ((size_t)(b * COUTc + c)) * NPOS + (size_t)n * Kk;
    const size_t kvbase = (((size_t)(b * COUTc + c)) * Nn + n) * (Kk * Ll);

    float m[20];
    float mx = -__builtin_inff();
    #pragma unroll
    for (int j = 0; j < 20; ++j) {
        float kv = kmem[kvbase + topi[j]];
        m[j] = qq[qbase + j] * kv;
        mx = fmaxf(mx, m[j]);
    }
    float ssum = 0.f;
    #pragma unroll
    for (int j = 0; j < 20; ++j) { m[j] = __expf(m[j] - mx); ssum += m[j]; }
    const float inv = 1.0f / ssum;

    float o = 0.f;
    #pragma unroll
    for (int j = 0; j < 20; ++j) o += m[j] * inv * vmem[kvbase + topi[j]];

    out[((size_t)(b * 2 * COUTc + COUTc + c)) * Nn + n] = o;
}

// ---------------------------------------------------------------------------
extern "C" void kernel_launch(void* const* d_in, const int* in_sizes, int n_in,
                              void* d_out, int out_size, void* d_ws, size_t ws_size,
                              hipStream_t stream)
{
    const float* x     = (const float*)d_in[0];   // (B,128,N,K)
    const float* k     = (const float*)d_in[1];   // (B,64,N,K,L)
    const float* v     = (const float*)d_in[2];   // (B,64,N,K,L)
    const float* s     = (const float*)d_in[3];   // (B,64,N,L)
    const float* abs_x = (const float*)d_in[4];   // (B,64,N,K)
    // d_in[5] = deg: unused by the reference
    const int*   idx   = (const int*)d_in[6];     // (B,N,K,L+1)
    const float* Wq    = (const float*)d_in[7];   // (64,64)
    const float* Wk    = (const float*)d_in[8];   // (64,128)
    const float* Wv    = (const float*)d_in[9];   // (64,128)

    // Output layout (flat, return order): out | k_query | v_query | score_query
    float* out   = (float*)d_out;                                   // 2,097,152
    float* kq    = out + (size_t)Bsz * 2 * COUTc * Nn;              // 20,971,520
    float* vq    = kq  + (size_t)Bsz * COUTc * NPOS;                // 20,971,520
    float* score = vq  + (size_t)Bsz * COUTc * NPOS;                //  1,048,576
    float* qq    = (float*)d_ws;                                    // q_query scratch

    // segment-sum target must start at zero (graph-capturable async memset)
    hipMemsetAsync(score, 0, (size_t)Bsz * COUTc * Nn * sizeof(float), stream);

    gemm_qkv_kernel<<<dim3(NPOS / 64, Bsz), 128, 0, stream>>>(x, abs_x, Wq, Wk, Wv,
                                                              kq, vq, qq);
    attn_query_kernel<<<dim3(Nn / 4, Bsz), 256, 0, stream>>>(qq, kq, vq, idx, out, score);
    attn_mem_kernel<<<(Bsz * COUTc * Nn) / 256, 256, 0, stream>>>(qq, k, v, s, idx, out);
}